// RNN_single_65730179498034
// MI455X (gfx1250) — compile-verified
//
#include <hip/hip_runtime.h>
#include <math.h>

#define T_STEPS 500
#define BATCH   64
#define NIN     64
#define NH      1024
#define NOUT    3
#define ALPHA   0.1f

typedef float v2f __attribute__((ext_vector_type(2)));
typedef float v8f __attribute__((ext_vector_type(8)));

// Packed fragment layouts: consecutive K pairs per float2.
//   A-side (activations/input), per K-pair row:  Ap[kr][m] = {A[m][2kr], A[m][2kr+1]}  (m = 0..63)
//   B-side (weights),          per K-pair row:  Bp[kr][n] = {W[n][2kr], W[n][2kr+1]}  (n = 0..NH-1)
// A lane (half, r) at k-group k4 needs row kr = 2*k4 + half -> ONE b64 load each side.

__device__ __forceinline__ v8f wmma4(float2 a, float2 b, v8f c) {
    v2f av; av.x = a.x; av.y = a.y;
    v2f bv; bv.x = b.x; bv.y = b.y;
    return __builtin_amdgcn_wmma_f32_16x16x4_f32(false, av, false, bv,
                                                 (short)0, c, false, false);
}

// float index of element (k, m) inside packed A-side buffer (BATCH columns)
__device__ __forceinline__ size_t apack_f(int k, int m) {
    return (size_t)(k >> 1) * (2 * BATCH) + 2 * m + (k & 1);
}

// ---------------------------------------------------------------------------
// Init: build packed operands + seed state.
// ---------------------------------------------------------------------------
__global__ __launch_bounds__(256) void rnn_init(
    const float* __restrict__ X, const float* __restrict__ hidden1,
    const float* __restrict__ Wih, const float* __restrict__ Whh,
    float2* __restrict__ WTp,    // [NH/2][NH]
    float2* __restrict__ WihTp,  // [NIN/2][NH]
    float2* __restrict__ XTp,    // [T][NIN/2][B]
    float* __restrict__ Xstate,  // [B][NH]
    float* __restrict__ RT0p)    // packed A-side, float view
{
    long id = (long)blockIdx.x * blockDim.x + threadIdx.x;
    const long nWhh = (long)(NH / 2) * NH;
    const long nWih = (long)(NIN / 2) * NH;
    const long nX   = (long)T_STEPS * (NIN / 2) * BATCH;
    const long nH   = (long)BATCH * NH;

    if (id < nWhh) {
        long kr = id / NH, n = id % NH;
        float2 v; v.x = Whh[n * NH + 2 * kr]; v.y = Whh[n * NH + 2 * kr + 1];
        WTp[kr * NH + n] = v;
        return;
    }
    id -= nWhh;
    if (id < nWih) {
        long kr = id / NH, n = id % NH;
        float2 v; v.x = Wih[n * NIN + 2 * kr]; v.y = Wih[n * NIN + 2 * kr + 1];
        WihTp[kr * NH + n] = v;
        return;
    }
    id -= nWih;
    if (id < nX) {
        long t = id / ((NIN / 2) * BATCH);
        long rem = id % ((NIN / 2) * BATCH);
        long kr = rem / BATCH, b = rem % BATCH;
        float2 v;
        v.x = X[(t * BATCH + b) * NIN + 2 * kr];
        v.y = X[(t * BATCH + b) * NIN + 2 * kr + 1];
        XTp[(t * (NIN / 2) + kr) * BATCH + b] = v;
        return;
    }
    id -= nX;
    if (id < nH) {
        long b = id / NH, n = id % NH;
        float x = hidden1[b * NH + n];
        Xstate[b * NH + n] = x;
        RT0p[apack_f((int)n, (int)b)] = tanhf(x);  // n is the K index next step
    }
}

// ---------------------------------------------------------------------------
// One recurrent step. 128 waves (16 blocks x 8 waves); each wave owns a
// 16(M) x 32(N) strip = 2 N-tiles, with 2 K-split accumulators per tile
// (4 independent WMMA chains). Loop body: 6 x b64 loads, 4 x wmma.
// ---------------------------------------------------------------------------
__global__ __launch_bounds__(256) void rnn_step(
    const float2* __restrict__ RTp,    // [NH/2][B]   packed prev activations
    const float2* __restrict__ XTp,    // [NIN/2][B]  packed input at t
    const float2* __restrict__ WTp,    // [NH/2][NH]  packed Whh^T
    const float2* __restrict__ WihTp,  // [NIN/2][NH] packed Wih^T
    float* __restrict__ Xstate,        // [B][NH]
    float* __restrict__ RTnextp,       // packed A-side, float view
    float* __restrict__ hid_t)         // [B][NH] hiddenl1 slice for t
{
    const int lane  = threadIdx.x & 31;
    const int wave  = threadIdx.x >> 5;
    const int tile  = blockIdx.x * 8 + wave;   // 0..127
    const int mt    = tile >> 5;               // 0..3   (batch tile)
    const int ntp   = tile & 31;               // 0..31  (pair of hidden tiles)
    const int half  = (lane >> 4) & 1;
    const int r     = lane & 15;
    const int mbase = mt * 16;

    const int aCol  = mbase + r;          // A-fragment column (m)
    const int bCol0 = ntp * 32 + r;       // B-fragment column, tile 0
    const int bCol1 = bCol0 + 16;         // B-fragment column, tile 1

    v8f acc00 = {0.f,0.f,0.f,0.f,0.f,0.f,0.f,0.f};  // ksplit0, ntile0
    v8f acc01 = acc00;                               // ksplit0, ntile1
    v8f acc10 = acc00;                               // ksplit1, ntile0
    v8f acc11 = acc00;                               // ksplit1, ntile1

    // ---- r1 @ Whh^T : 256 k-groups, processed two per iteration ----
    for (int k4 = 0; k4 < NH / 4; k4 += 2) {
        int rowE = 2 * k4 + half;          // K-pair row, even k-group
        int rowO = rowE + 2;               // K-pair row, odd  k-group
        float2 aE  = RTp[(size_t)rowE * BATCH + aCol];
        float2 aO  = RTp[(size_t)rowO * BATCH + aCol];
        float2 b0E = WTp[(size_t)rowE * NH + bCol0];
        float2 b1E = WTp[(size_t)rowE * NH + bCol1];
        float2 b0O = WTp[(size_t)rowO * NH + bCol0];
        float2 b1O = WTp[(size_t)rowO * NH + bCol1];
        acc00 = wmma4(aE, b0E, acc00);
        acc01 = wmma4(aE, b1E, acc01);
        acc10 = wmma4(aO, b0O, acc10);
        acc11 = wmma4(aO, b1O, acc11);
    }
    // ---- x_in @ Wih^T : 16 k-groups ----
    for (int k4 = 0; k4 < NIN / 4; k4 += 2) {
        int rowE = 2 * k4 + half;
        int rowO = rowE + 2;
        float2 aE  = XTp[(size_t)rowE * BATCH + aCol];
        float2 aO  = XTp[(size_t)rowO * BATCH + aCol];
        float2 b0E = WihTp[(size_t)rowE * NH + bCol0];
        float2 b1E = WihTp[(size_t)rowE * NH + bCol1];
        float2 b0O = WihTp[(size_t)rowO * NH + bCol0];
        float2 b1O = WihTp[(size_t)rowO * NH + bCol1];
        acc00 = wmma4(aE, b0E, acc00);
        acc01 = wmma4(aE, b1E, acc01);
        acc10 = wmma4(aO, b0O, acc10);
        acc11 = wmma4(aO, b1O, acc11);
    }

    v8f acc0 = acc00 + acc10;   // ntile 0
    v8f acc1 = acc01 + acc11;   // ntile 1

    // ---- leaky update + tanh + stores (C layout: VGPR i -> M = i+8*half) ----
#pragma unroll
    for (int i = 0; i < 8; ++i) {
        int m = mbase + i + half * 8;
        {
            size_t idx = (size_t)m * NH + bCol0;
            float xo = Xstate[idx];
            float xn = xo + ALPHA * (acc0[i] - xo);
            float rr = tanhf(xn);
            Xstate[idx] = xn;
            hid_t[idx]  = rr;
            RTnextp[apack_f(bCol0, m)] = rr;
        }
        {
            size_t idx = (size_t)m * NH + bCol1;
            float xo = Xstate[idx];
            float xn = xo + ALPHA * (acc1[i] - xo);
            float rr = tanhf(xn);
            Xstate[idx] = xn;
            hid_t[idx]  = rr;
            RTnextp[apack_f(bCol1, m)] = rr;
        }
    }
}

// ---------------------------------------------------------------------------
// Output head over all (t,b): one wave per (t,b).
// ---------------------------------------------------------------------------
__global__ __launch_bounds__(256) void rnn_out(
    const float* __restrict__ hid,     // [T][B][NH]
    const float* __restrict__ perturb, // [T][B][NOUT][NOUT]
    const float* __restrict__ Wout,    // [NOUT][NH]
    const float* __restrict__ bout,    // [NOUT]
    float* __restrict__ outv,          // [T][B][NOUT]
    float* __restrict__ outp)          // [T][B][NOUT]
{
    const int lane = threadIdx.x & 31;
    const int wave = threadIdx.x >> 5;
    const long tb  = (long)blockIdx.x * 8 + wave;
    const float* h = hid + tb * NH;

    float s0 = 0.f, s1 = 0.f, s2 = 0.f;
    for (int k = lane; k < NH; k += 32) {
        float hv = h[k];
        s0 += hv * Wout[0 * NH + k];
        s1 += hv * Wout[1 * NH + k];
        s2 += hv * Wout[2 * NH + k];
    }
#pragma unroll
    for (int off = 16; off > 0; off >>= 1) {
        s0 += __shfl_xor(s0, off, 32);
        s1 += __shfl_xor(s1, off, 32);
        s2 += __shfl_xor(s2, off, 32);
    }
    if (lane == 0) {
        float o0 = s0 + bout[0];
        float o1 = s1 + bout[1];
        float o2 = s2 + bout[2];
        const float* p = perturb + tb * NOUT * NOUT;
        outv[tb * NOUT + 0] = o0;
        outv[tb * NOUT + 1] = o1;
        outv[tb * NOUT + 2] = o2;
        outp[tb * NOUT + 0] = p[0] * o0 + p[1] * o1 + p[2] * o2;
        outp[tb * NOUT + 1] = p[3] * o0 + p[4] * o1 + p[5] * o2;
        outp[tb * NOUT + 2] = p[6] * o0 + p[7] * o1 + p[8] * o2;
    }
}

// ---------------------------------------------------------------------------
extern "C" void kernel_launch(void* const* d_in, const int* in_sizes, int n_in,
                              void* d_out, int out_size, void* d_ws, size_t ws_size,
                              hipStream_t stream)
{
    const float* X       = (const float*)d_in[0];
    const float* perturb = (const float*)d_in[1];
    const float* hidden1 = (const float*)d_in[2];
    const float* Wih     = (const float*)d_in[3];
    const float* Whh     = (const float*)d_in[4];
    const float* Wout    = (const float*)d_in[5];
    const float* bout    = (const float*)d_in[6];

    float* outv = (float*)d_out;
    float* outp = outv + (size_t)T_STEPS * BATCH * NOUT;
    float* hid  = outp + (size_t)T_STEPS * BATCH * NOUT;

    float* ws      = (float*)d_ws;
    float2* WTp    = (float2*)ws;  ws += (size_t)NH * NH;                    // 4 MB
    float2* WihTp  = (float2*)ws;  ws += (size_t)NIN * NH;                   // 256 KB
    float2* XTp    = (float2*)ws;  ws += (size_t)T_STEPS * NIN * BATCH;      // 8 MB
    float* Xstate  = ws;           ws += (size_t)BATCH * NH;                 // 256 KB
    float* RT0p    = ws;           ws += (size_t)NH * BATCH;                 // 256 KB
    float* RT1p    = ws;           ws += (size_t)NH * BATCH;                 // 256 KB

    long initN = (long)(NH / 2) * NH + (long)(NIN / 2) * NH +
                 (long)T_STEPS * (NIN / 2) * BATCH + (long)BATCH * NH;
    rnn_init<<<(unsigned)((initN + 255) / 256), 256, 0, stream>>>(
        X, hidden1, Wih, Whh, WTp, WihTp, XTp, Xstate, RT0p);

    for (int t = 0; t < T_STEPS; ++t) {
        const float* RTprev = (t & 1) ? RT1p : RT0p;
        float*       RTnext = (t & 1) ? RT0p : RT1p;
        rnn_step<<<16, 256, 0, stream>>>(
            (const float2*)RTprev,
            XTp + (size_t)t * (NIN / 2) * BATCH,
            WTp, WihTp, Xstate,
            RTnext,
            hid + (size_t)t * BATCH * NH);
    }

    rnn_out<<<(unsigned)((T_STEPS * BATCH) / 8), 256, 0, stream>>>(
        hid, perturb, Wout, bout, outv, outp);
}